// JointCCSA_28767690948905
// MI455X (gfx1250) — compile-verified
//
#include <hip/hip_runtime.h>

typedef __attribute__((ext_vector_type(2))) float v2f;
typedef __attribute__((ext_vector_type(8))) float v8f;

#define BS   4096
#define DIM  512
#define KC   64          // K chunk staged in LDS
#define TILE 64          // super-tile edge (4x4 of 16x16 WMMA tiles)
#define LDW  (KC + 4)    // padded LDS row stride (68 dwords -> conflict-free)

__global__ __launch_bounds__(256)
void ccsa_rowsq_kernel(const float* __restrict__ X, float* __restrict__ sq) {
    const int wid  = threadIdx.x >> 5;
    const int lane = threadIdx.x & 31;
    const int row  = blockIdx.x * 8 + wid;
    const float* xr = X + (size_t)row * DIM;
    float s = 0.f;
#pragma unroll
    for (int q = 0; q < DIM / 32; ++q) {
        float v = xr[lane + 32 * q];
        s = fmaf(v, v, s);
    }
#pragma unroll
    for (int m = 16; m >= 1; m >>= 1) s += __shfl_xor(s, m, 32);
    if (lane == 0) sq[row] = s;
}

__global__ __launch_bounds__(256)
void ccsa_main_kernel(const float* __restrict__ X,
                      const float* __restrict__ sq,
                      const int*   __restrict__ dsv,
                      const int*   __restrict__ yv,
                      const int*   __restrict__ ncls,
                      const int*   __restrict__ ndom,
                      float*       __restrict__ out) {
    const int Iblk = blockIdx.y;
    const int Jblk = blockIdx.x;
    if (Iblk > Jblk) return;            // symmetry: dist(i,j)==dist(j,i); both
                                        // orientations handled in the epilogue
    __shared__ float As[TILE][LDW];
    __shared__ float Bs[TILE][LDW];
    __shared__ float sqI[TILE], sqJ[TILE];
    __shared__ int   yI[TILE],  yJ[TILE], dsI[TILE], dsJ[TILE];
    __shared__ float red[2][8];

    const int Ibase = Iblk * TILE;
    const int Jbase = Jblk * TILE;
    const int tid  = threadIdx.x;
    const int lane = tid & 31;
    const int wid  = tid >> 5;

    // stage per-row metadata for this super-tile
    if (tid < TILE) {
        sqI[tid] = sq[Ibase + tid];
        yI[tid]  = yv[Ibase + tid];
        dsI[tid] = dsv[Ibase + tid];
    } else if (tid < 2 * TILE) {
        int t = tid - TILE;
        sqJ[t] = sq[Jbase + t];
        yJ[t]  = yv[Jbase + t];
        dsJ[t] = dsv[Jbase + t];
    }

    // subtile assignment: wave handles (ti, tj0) and (ti, tj0+1) -> A reuse
    const int ti  = wid >> 1;
    const int tj0 = (wid & 1) * 2;
    const int ml  = lane & 15;          // M (A) / N (B,C) low index
    const int hi  = lane >> 4;          // lane-half select

    v8f acc0 = {0.f, 0.f, 0.f, 0.f, 0.f, 0.f, 0.f, 0.f};
    v8f acc1 = acc0;

    const int ldrow = tid >> 2;         // 0..63
    const int ldcol = (tid & 3) * 16;   // 0,16,32,48  (4x float4 = 64B contiguous)

    // raw LDS byte offsets for async-to-LDS (generic LDS addr: low 32b = LDS offset)
    const unsigned ldsA = (unsigned)(uintptr_t)&As[ldrow][ldcol];
    const unsigned ldsB = (unsigned)(uintptr_t)&Bs[ldrow][ldcol];

    for (int kc = 0; kc < DIM; kc += KC) {
        __syncthreads();                // LDS reuse guard
        // CDNA5 async cache->LDS DMA: no VGPR bounce, tracked by ASYNCcnt.
        // INST_OFFSET advances global and LDS addresses in lockstep, which
        // matches this thread's contiguous 64B share of the tile.
        const float* ga = X + (size_t)(Ibase + ldrow) * DIM + kc + ldcol;
        const float* gb = X + (size_t)(Jbase + ldrow) * DIM + kc + ldcol;
        asm volatile(
            "global_load_async_to_lds_b128 %0, %2, off\n\t"
            "global_load_async_to_lds_b128 %0, %2, off offset:16\n\t"
            "global_load_async_to_lds_b128 %0, %2, off offset:32\n\t"
            "global_load_async_to_lds_b128 %0, %2, off offset:48\n\t"
            "global_load_async_to_lds_b128 %1, %3, off\n\t"
            "global_load_async_to_lds_b128 %1, %3, off offset:16\n\t"
            "global_load_async_to_lds_b128 %1, %3, off offset:32\n\t"
            "global_load_async_to_lds_b128 %1, %3, off offset:48"
            :: "v"(ldsA), "v"(ldsB), "v"(ga), "v"(gb) : "memory");
        asm volatile("s_wait_asynccnt 0x0" ::: "memory");
        __syncthreads();

#pragma unroll
        for (int k = 0; k < KC / 4; ++k) {
            const int kidx = 4 * k + 2 * hi;   // A/B fragment: lane holds K = kidx, kidx+1
            v2f a  = *reinterpret_cast<const v2f*>(&As[ti * 16 + ml][kidx]);
            v2f b0 = *reinterpret_cast<const v2f*>(&Bs[tj0 * 16 + ml][kidx]);
            v2f b1 = *reinterpret_cast<const v2f*>(&Bs[tj0 * 16 + 16 + ml][kidx]);
            acc0 = __builtin_amdgcn_wmma_f32_16x16x4_f32(false, a, false, b0,
                                                         (short)0, acc0, false, false);
            acc1 = __builtin_amdgcn_wmma_f32_16x16x4_f32(false, a, false, b1,
                                                         (short)0, acc1, false, false);
        }
    }

    // epilogue: dist + masked sums, both pair orientations
    const bool offdiag = (Iblk != Jblk);
    float sa = 0.f, sl = 0.f;
#pragma unroll
    for (int t = 0; t < 2; ++t) {
        const int tj = tj0 + t;
        const float sj = sqJ[tj * 16 + ml];
        const int   yj = yJ[tj * 16 + ml];
        const int   dj = dsJ[tj * 16 + ml];
        v8f acc = (t == 0) ? acc0 : acc1;
#pragma unroll
        for (int r = 0; r < 8; ++r) {
            const int mi = ti * 16 + r + 8 * hi;   // C layout: VGPR r -> M = r + 8*hi
            const float si = sqI[mi];
            const int   yi = yI[mi];
            const int   di = dsI[mi];
            float d2 = si + sj - 2.f * acc[r];
            d2 = fmaxf(d2, 0.f);
            const float dist = sqrtf(d2);
            const float hinge = fmaxf(0.f, 1.f - dist);
            // orientation (i, j)
            if (di < dj) {
                if (yi == yj) sa += dist;
                if (yi <  yj) sl += hinge;
            }
            // orientation (j, i) — only for off-diagonal tiles
            if (offdiag && (dj < di)) {
                if (yj == yi) sa += dist;
                if (yj <  yi) sl += hinge;
            }
        }
    }

    // wave reduction then cross-wave via LDS
#pragma unroll
    for (int m = 16; m >= 1; m >>= 1) {
        sa += __shfl_xor(sa, m, 32);
        sl += __shfl_xor(sl, m, 32);
    }
    if (lane == 0) { red[0][wid] = sa; red[1][wid] = sl; }
    __syncthreads();
    if (tid == 0) {
        float tsa = 0.f, tsl = 0.f;
#pragma unroll
        for (int w = 0; w < 8; ++w) { tsa += red[0][w]; tsl += red[1][w]; }
        const int nc = *ncls;
        const int nd = *ndom;
        const float npair_d = (float)(nd * (nd - 1) / 2);
        const float n_sa = (float)nc * npair_d;
        const float n_s  = (float)(nc * (nc - 1) / 2) * npair_d;
        atomicAdd(&out[0], 0.5f * tsa / n_sa);
        atomicAdd(&out[1], 0.5f * tsl / n_s);
    }
}

extern "C" void kernel_launch(void* const* d_in, const int* in_sizes, int n_in,
                              void* d_out, int out_size, void* d_ws, size_t ws_size,
                              hipStream_t stream) {
    (void)in_sizes; (void)n_in; (void)out_size; (void)ws_size;
    const float* X    = (const float*)d_in[0];
    const int*   dsv  = (const int*)d_in[1];
    const int*   yv   = (const int*)d_in[2];
    const int*   ncls = (const int*)d_in[3];
    const int*   ndom = (const int*)d_in[4];
    float*       out  = (float*)d_out;
    float*       sq   = (float*)d_ws;          // 4096 floats of scratch

    hipMemsetAsync(d_out, 0, 2 * sizeof(float), stream);

    ccsa_rowsq_kernel<<<BS / 8, 256, 0, stream>>>(X, sq);

    dim3 grid(BS / TILE, BS / TILE);           // upper-triangular blocks do the work
    ccsa_main_kernel<<<grid, 256, 0, stream>>>(X, sq, dsv, yv, ncls, ndom, out);
}